// BDC_Representation_12945031430164
// MI455X (gfx1250) — compile-verified
//
#include <hip/hip_runtime.h>
#include <hip/hip_bf16.h>
#include <math.h>

// Problem constants
#define BATCH 64
#define MTOK  197
#define DDIM  768
#define KPAD  200          // K padded to multiple of 40 (LDS chunk)
#define KCHUNK 40
#define TRI   295296       // D*(D+1)/2
#define NBI   6            // D/128 macro-tile rows
#define NBJ   12           // D/64  macro-tile cols
#define NPAIR 42           // sum_{bi} (12 - 2*bi)
#define INV2M (1.0f/394.0f)
#define INVSQRTD 0.03608439182435161f   // 1/sqrt(768)
#define INVD  (1.0f/768.0f)

#define ASTRIDE 136        // A slab row stride (floats): 2-row bank shift = 16
#define BSTRIDE 80         // B slab row stride (floats): 2-row bank shift = 32

typedef __attribute__((ext_vector_type(2))) float v2f;
typedef __attribute__((ext_vector_type(8))) float v8f;

static __device__ __forceinline__ v8f wmma_f32_16x16x4(v2f a, v2f b, v8f c) {
  // (neg_a, A, neg_b, B, c_mod, C, reuse_a, reuse_b)
  return __builtin_amdgcn_wmma_f32_16x16x4_f32(false, a, false, b, (short)0, c,
                                               false, false);
}

// ---------------------------------------------------------------------------
// K1: per-batch attention scores + softmax + weighted tokens z = w ⊙ x
// grid = BATCH blocks, 256 threads (8 waves of 32)
// z layout: [BATCH][KPAD][DDIM], rows >= MTOK zeroed.
// ---------------------------------------------------------------------------
__global__ void k1_softmax_z(const float* __restrict__ x_g,
                             const float* __restrict__ x,
                             float* __restrict__ z) {
  __shared__ float xs[DDIM];
  __shared__ float sm[256];
  __shared__ float red[256];

  const int b  = blockIdx.x;
  const int t  = threadIdx.x;
  const int wv = t >> 5;
  const int ln = t & 31;
  const size_t xb = (size_t)b * MTOK * DDIM;
  const size_t zb = (size_t)b * KPAD * DDIM;

  for (int d = t; d < DDIM; d += 256) xs[d] = x_g[(size_t)b * DDIM + d];
  __syncthreads();

  // wave-cooperative dot products: wave wv handles tokens wv, wv+8, ...
  for (int m = wv; m < MTOK; m += 8) {
    float p = 0.f;
    const float* xr = x + xb + (size_t)m * DDIM;
    for (int d = ln; d < DDIM; d += 32) p += xs[d] * xr[d];
    for (int off = 16; off > 0; off >>= 1) p += __shfl_xor(p, off);
    if (ln == 0) sm[m] = p * INVSQRTD;
  }
  __syncthreads();

  // block softmax over MTOK scores
  float sc = (t < MTOK) ? sm[t] : -3.0e38f;
  red[t] = sc;
  __syncthreads();
  for (int s = 128; s > 0; s >>= 1) {
    if (t < s) red[t] = fmaxf(red[t], red[t + s]);
    __syncthreads();
  }
  float mx = red[0];
  __syncthreads();
  float p = (t < MTOK) ? expf(sc - mx) : 0.f;
  red[t] = p;
  __syncthreads();
  for (int s = 128; s > 0; s >>= 1) {
    if (t < s) red[t] += red[t + s];
    __syncthreads();
  }
  float tot = red[0];
  __syncthreads();
  if (t < MTOK) sm[t] = p / tot;
  __syncthreads();

  // write z (coalesced per wave), zero pad rows
  for (int row = wv; row < KPAD; row += 8) {
    float* zr = z + zb + (size_t)row * DDIM;
    if (row < MTOK) {
      const float wm = sm[row];
      const float* xr = x + xb + (size_t)row * DDIM;
      for (int d = ln; d < DDIM; d += 32) zr[d] = wm * xr[d];
    } else {
      for (int d = ln; d < DDIM; d += 32) zr[d] = 0.f;
    }
  }
}

// ---------------------------------------------------------------------------
// K2: diag[b,i] = sum_m z[b,m,i]^2 / (2M)
// ---------------------------------------------------------------------------
__global__ void k2_diag(const float* __restrict__ z, float* __restrict__ diag) {
  const int idx = blockIdx.x * 256 + threadIdx.x;
  if (idx >= BATCH * DDIM) return;
  const int b = idx / DDIM;
  const int i = idx - b * DDIM;
  const float* zp = z + (size_t)b * KPAD * DDIM + i;
  float s = 0.f;
  for (int m = 0; m < MTOK; ++m) {
    float v = zp[(size_t)m * DDIM];
    s += v * v;
  }
  diag[idx] = s * INV2M;
}

// ---------------------------------------------------------------------------
// K3: WMMA SYRK G = z^T z / (2M) on 128x64 macro tiles covering the upper
//     triangle, fused dcov = sqrt(clip(diag_i + diag_j - 2G, 0) + 1e-5),
//     packed upper-tri write to out.
// grid = (NPAIR, BATCH), 256 threads = 8 waves.
// wave w: 2x2 subtiles at rows (w>>1)*2+{0,1}, cols (w&1)*2+{0,1}.
// ---------------------------------------------------------------------------
__global__ void __launch_bounds__(256)
k3_wmma_dcov(const float* __restrict__ z, const float* __restrict__ diag,
             float* __restrict__ out) {
  __shared__ float As[KCHUNK][ASTRIDE];   // 128-wide slab (rows of C)
  __shared__ float Bs[KCHUNK][BSTRIDE];   // 64-wide slab (cols of C)

  // map flat index -> (bi, bj): bi in [0,6), bj in [2*bi, 12)
  int u = blockIdx.x;
  int bi = 0;
  while (u >= NBJ - 2 * bi) { u -= NBJ - 2 * bi; ++bi; }
  const int bj = 2 * bi + u;
  const int b  = blockIdx.y;

  const int i0 = bi * 128;
  const int j0 = bj * 64;
  const int t    = threadIdx.x;
  const int wv   = t >> 5;
  const int ln   = t & 31;
  const int ln15 = ln & 15;
  const int half = ln >> 4;            // 0: lanes 0-15, 1: lanes 16-31
  const int tib  = (wv >> 1) * 2;      // subtile row base 0,2,4,6
  const int tjb  = (wv & 1) * 2;       // subtile col base 0,2

  const float* zb = z + (size_t)b * KPAD * DDIM;

  v8f acc00 = {}, acc01 = {}, acc10 = {}, acc11 = {};

  for (int k0 = 0; k0 < KPAD; k0 += KCHUNK) {
    // stage A slab: 40 x 128 (5120 elems, 20/thread), coalesced
    for (int e = 0; e < 20; ++e) {
      const int l   = t + 256 * e;
      const int row = l >> 7;
      const int col = l & 127;
      As[row][col] = zb[(size_t)(k0 + row) * DDIM + i0 + col];
    }
    // stage B slab: 40 x 64 (2560 elems, 10/thread), coalesced
    for (int e = 0; e < 10; ++e) {
      const int l   = t + 256 * e;
      const int row = l >> 6;
      const int col = l & 63;
      Bs[row][col] = zb[(size_t)(k0 + row) * DDIM + j0 + col];
    }
    __syncthreads();

    #pragma unroll
    for (int ks = 0; ks < KCHUNK / 4; ++ks) {
      const int kk = ks * 4 + 2 * half;
      v2f a0, a1, b0v, b1v;
      a0.x  = As[kk    ][tib * 16 + ln15];
      a0.y  = As[kk + 1][tib * 16 + ln15];
      a1.x  = As[kk    ][(tib + 1) * 16 + ln15];
      a1.y  = As[kk + 1][(tib + 1) * 16 + ln15];
      b0v.x = Bs[kk    ][tjb * 16 + ln15];
      b0v.y = Bs[kk + 1][tjb * 16 + ln15];
      b1v.x = Bs[kk    ][(tjb + 1) * 16 + ln15];
      b1v.y = Bs[kk + 1][(tjb + 1) * 16 + ln15];
      acc00 = wmma_f32_16x16x4(a0, b0v, acc00);
      acc01 = wmma_f32_16x16x4(a0, b1v, acc01);
      acc10 = wmma_f32_16x16x4(a1, b0v, acc10);
      acc11 = wmma_f32_16x16x4(a1, b1v, acc11);
    }
    __syncthreads();
  }

  // epilogue: acc -> dcov, packed upper-tri scatter
  const float* dgb = diag + (size_t)b * DDIM;
  const size_t obase = (size_t)b * TRI;
  #pragma unroll
  for (int r = 0; r < 2; ++r) {
    #pragma unroll
    for (int c = 0; c < 2; ++c) {
      v8f acc = (r == 0) ? ((c == 0) ? acc00 : acc01)
                         : ((c == 0) ? acc10 : acc11);
      const int gj = j0 + (tjb + c) * 16 + ln15;
      const float dj = dgb[gj];
      #pragma unroll
      for (int v = 0; v < 8; ++v) {
        const int gi = i0 + (tib + r) * 16 + v + 8 * half;
        const float g  = acc[v] * INV2M;
        const float di = dgb[gi];
        const float dc = sqrtf(fmaxf(di + dj - 2.0f * g, 0.0f) + 1e-5f);
        if (gi <= gj) {
          const size_t off = (size_t)gi * DDIM - (size_t)gi * (gi - 1) / 2;
          out[obase + off + (gj - gi)] = dc;
        }
      }
    }
  }
}

// ---------------------------------------------------------------------------
// K4: rowsum[b,i] = sum_j dcov[b,i,j] (dcov symmetric, read packed from out)
// grid = (DDIM, BATCH), 256 threads, tree reduction.
// ---------------------------------------------------------------------------
__global__ void k4_rowsum(const float* __restrict__ out,
                          float* __restrict__ rowsum) {
  __shared__ float red[256];
  const int i = blockIdx.x;
  const int b = blockIdx.y;
  const int t = threadIdx.x;
  const size_t base = (size_t)b * TRI;
  float s = 0.f;
  for (int j = t; j < DDIM; j += 256) {
    size_t idx;
    if (j >= i)
      idx = base + (size_t)i * DDIM - (size_t)i * (i - 1) / 2 + (j - i);
    else
      idx = base + (size_t)j * DDIM - (size_t)j * (j - 1) / 2 + (i - j);
    s += out[idx];
  }
  red[t] = s;
  __syncthreads();
  for (int st = 128; st > 0; st >>= 1) {
    if (t < st) red[t] += red[t + st];
    __syncthreads();
  }
  if (t == 0) rowsum[(size_t)b * DDIM + i] = red[0];
}

// ---------------------------------------------------------------------------
// K5: gm[b] = sum_i rowsum[b,i] / (D*D)
// ---------------------------------------------------------------------------
__global__ void k5_grandmean(const float* __restrict__ rowsum,
                             float* __restrict__ gm) {
  __shared__ float red[256];
  const int b = blockIdx.x;
  const int t = threadIdx.x;
  float s = 0.f;
  for (int i = t; i < DDIM; i += 256) s += rowsum[(size_t)b * DDIM + i];
  red[t] = s;
  __syncthreads();
  for (int st = 128; st > 0; st >>= 1) {
    if (t < st) red[t] += red[t + st];
    __syncthreads();
  }
  if (t == 0) gm[b] = red[0] * (INVD * INVD);
}

// ---------------------------------------------------------------------------
// K6: in-place double centering of packed out:
//     out[b, i, j] -= rm[i] + rm[j] - gm   (j >= i packed)
// grid = (DDIM, BATCH)
// ---------------------------------------------------------------------------
__global__ void k6_center(float* __restrict__ out,
                          const float* __restrict__ rowsum,
                          const float* __restrict__ gm) {
  const int i = blockIdx.x;
  const int b = blockIdx.y;
  const int t = threadIdx.x;
  const float g   = gm[b];
  const float rmi = rowsum[(size_t)b * DDIM + i] * INVD;
  const size_t base =
      (size_t)b * TRI + (size_t)i * DDIM - (size_t)i * (i - 1) / 2;
  const int len = DDIM - i;
  for (int e = t; e < len; e += 256) {
    const int j = i + e;
    const float rmj = rowsum[(size_t)b * DDIM + j] * INVD;
    out[base + e] = out[base + e] - rmi - rmj + g;
  }
}

// ---------------------------------------------------------------------------
extern "C" void kernel_launch(void* const* d_in, const int* in_sizes, int n_in,
                              void* d_out, int out_size, void* d_ws,
                              size_t ws_size, hipStream_t stream) {
  const float* x_g = (const float*)d_in[0];   // [B, D]
  const float* x   = (const float*)d_in[1];   // [B, M, D]
  float* out = (float*)d_out;                 // [B, TRI]

  float* ws = (float*)d_ws;
  float* z      = ws;                                   // B*KPAD*D
  float* diag   = z + (size_t)BATCH * KPAD * DDIM;      // B*D
  float* rowsum = diag + (size_t)BATCH * DDIM;          // B*D
  float* gm     = rowsum + (size_t)BATCH * DDIM;        // B

  k1_softmax_z<<<dim3(BATCH), dim3(256), 0, stream>>>(x_g, x, z);

  const int nvec = BATCH * DDIM;
  k2_diag<<<dim3((nvec + 255) / 256), dim3(256), 0, stream>>>(z, diag);

  k3_wmma_dcov<<<dim3(NPAIR, BATCH), dim3(256), 0, stream>>>(z, diag, out);

  k4_rowsum<<<dim3(DDIM, BATCH), dim3(256), 0, stream>>>(out, rowsum);

  k5_grandmean<<<dim3(BATCH), dim3(256), 0, stream>>>(rowsum, gm);

  k6_center<<<dim3(DDIM, BATCH), dim3(256), 0, stream>>>(out, rowsum, gm);
}